// GatherNodes_58256936403575
// MI455X (gfx1250) — compile-verified
//
#include <hip/hip_runtime.h>

// GatherNodes: out[e] = concat(nodes[index[0][e]], nodes[index[1][e]])
// nodes: [N, 64] f32  (25.6 MB -> resident in MI455X's 192 MB L2)
// index: [2, M] i32
// out:   [M, 128] f32 (640 MB streamed -> non-temporal stores)
//
// Wave32 mapping: one wave per edge.
//   lanes 0..15  -> float4 slots 0..15 of receiver row (x_i)
//   lanes 16..31 -> float4 slots 0..15 of sender row   (x_j)
// Each lane: 1x b32 index load, 1x b128 node load (L2-hot),
//            1x b128 non-temporal store (contiguous 512B per wave).

// Native clang vector type: required by __builtin_nontemporal_store and
// lowers to single b128 vmem ops.
typedef float v4f __attribute__((ext_vector_type(4)));

__global__ void __launch_bounds__(256)
GatherNodes_58256936403575_kernel(const v4f* __restrict__ nodes,   // [N, 16] as v4f
                                  const int* __restrict__ index,   // [2, M]
                                  v4f*       __restrict__ out,     // [M, 32] as v4f
                                  int M)
{
    const int g    = blockIdx.x * blockDim.x + threadIdx.x;
    const int e    = g >> 5;        // edge id (one wave32 per edge)
    const int slot = g & 31;        // float4 slot within the 128-float output row
    if (e >= M) return;

    const int part = slot >> 4;     // 0 = receiver (index[0]), 1 = sender (index[1])
    const int sub  = slot & 15;     // float4 within the 64-float node row

    // Broadcast within each half-wave: 16 lanes hit the same 4B index word.
    const int src = index[part * M + e];

    // 256B fully-coalesced burst per half-wave from one node row; regular
    // temporal hint keeps the 25.6 MB table hot in L2 across all 2.5M gathers.
    const v4f v = nodes[(size_t)src * 16 + (size_t)sub];

    // 512B contiguous aligned row per wave; non-temporal so the 640 MB output
    // stream does not evict the nodes table from L2.
    __builtin_nontemporal_store(v, &out[(size_t)e * 32 + (size_t)slot]);
}

extern "C" void kernel_launch(void* const* d_in, const int* in_sizes, int n_in,
                              void* d_out, int out_size, void* d_ws, size_t ws_size,
                              hipStream_t stream) {
    const v4f* nodes = (const v4f*)d_in[0];
    const int* index = (const int*)d_in[1];
    v4f*       out   = (v4f*)d_out;

    const int M = in_sizes[1] / 2;          // index is [2, M]
    const long long total_threads = (long long)M * 32;
    const int block = 256;                  // 8 wave32s per block
    const int grid  = (int)((total_threads + block - 1) / block);

    GatherNodes_58256936403575_kernel<<<grid, block, 0, stream>>>(nodes, index, out, M);
}